// GNNModel_50680614092805
// MI455X (gfx1250) — compile-verified
//
#include <hip/hip_runtime.h>
#include <hip/hip_bf16.h>

#define NND 200000
#define NE  600000
#define FIN 165
#define FHID 128
#define FOUT 2

typedef float v2f __attribute__((ext_vector_type(2)));
typedef float v8f __attribute__((ext_vector_type(8)));

// ---------------------------------------------------------------------------
// 0) zero a float4 region (deg + agg1 + agg2 must start at 0 every call)
// ---------------------------------------------------------------------------
__global__ void zero_f4(float4* __restrict__ p, size_t n4) {
    size_t stride = (size_t)gridDim.x * blockDim.x;
    for (size_t i = (size_t)blockIdx.x * blockDim.x + threadIdx.x; i < n4; i += stride)
        p[i] = make_float4(0.f, 0.f, 0.f, 0.f);
}

// ---------------------------------------------------------------------------
// 1) degree count: deg[dst] += 1 for every edge (self-loop added later)
// ---------------------------------------------------------------------------
__global__ void degree_kernel(const int* __restrict__ ei, float* __restrict__ deg) {
    int e = blockIdx.x * blockDim.x + threadIdx.x;
    if (e < NE) atomicAdd(&deg[ei[NE + e]], 1.0f);
}

// deg -> dinv = rsqrt(deg + 1)   (+1 = self loop; always > 0)
__global__ void dinv_kernel(float* __restrict__ deg) {
    int n = blockIdx.x * blockDim.x + threadIdx.x;
    if (n < NND) deg[n] = rsqrtf(deg[n] + 1.0f);
}

// ---------------------------------------------------------------------------
// 2) GEMM1: h1[N,128] = x[N,165] @ W1[165,128]  via V_WMMA_F32_16X16X4_F32
//    One wave per 16x16 tile. 12500 row-tiles x 8 col-tiles = 100000 waves.
//    A layout (ISA 16x4 f32): lanes 0-15 -> K={k0,k0+1}, lanes 16-31 -> K={k0+2,k0+3}
//    C layout: VGPR i -> M = i (lanes 0-15) / i+8 (lanes 16-31), N = lane&15
//    165 = 41*4 + 1: 41 branch-free main iterations + one cndmask-masked tail
//    (keeps EXEC all-ones around every WMMA, no divergent loads in the loop).
// ---------------------------------------------------------------------------
__global__ void gemm1_wmma(const float* __restrict__ x,
                           const float* __restrict__ W1,
                           float* __restrict__ h1) {
    const int lane = threadIdx.x & 31;
    const int tile = blockIdx.x * (blockDim.x >> 5) + (threadIdx.x >> 5);
    const int rowTile = tile >> 3;   // 0..12499
    const int colTile = tile & 7;    // 0..7
    const int row0 = rowTile << 4;
    const int col0 = colTile << 4;
    const int m  = lane & 15;
    const int hi = lane >> 4;        // 0 for lanes 0-15, 1 for lanes 16-31
    const int col = col0 + m;

    v8f acc = {0.f, 0.f, 0.f, 0.f, 0.f, 0.f, 0.f, 0.f};
    const float* xrow = x + (size_t)(row0 + m) * FIN;

    // main loop: K = 0..163, fully in-bounds, no masking
    for (int k0 = 0; k0 < FIN - 4; k0 += 4) {
        const int kb = k0 + (hi << 1);
        v2f a, b;
        a.x = xrow[kb];
        a.y = xrow[kb + 1];
        b.x = W1[(size_t)kb * FHID + col];
        b.y = W1[(size_t)(kb + 1) * FHID + col];
        acc = __builtin_amdgcn_wmma_f32_16x16x4_f32(
            /*neg_a=*/false, a, /*neg_b=*/false, b,
            /*c_mod=*/(short)0, acc, /*reuse_a=*/false, /*reuse_b=*/false);
    }

    // tail: K = 164 only (valid for the low half-wave's first component).
    // Load the valid element in ALL lanes (address in-bounds everywhere),
    // zero the upper half-wave contribution via select -> no exec divergence.
    {
        const float a164 = xrow[FIN - 1];
        const float b164 = W1[(size_t)(FIN - 1) * FHID + col];
        v2f a, b;
        a.x = hi ? 0.f : a164;
        a.y = 0.f;
        b.x = hi ? 0.f : b164;
        b.y = 0.f;
        acc = __builtin_amdgcn_wmma_f32_16x16x4_f32(
            false, a, false, b, (short)0, acc, false, false);
    }

#pragma unroll
    for (int i = 0; i < 8; ++i)
        h1[(size_t)(row0 + i + 8 * hi) * FHID + col] = acc[i];
}

// ---------------------------------------------------------------------------
// 3) edge scatter layer 1: agg1[dst,:] += h1[src,:] * dinv[src]*dinv[dst]
//    one wave32 per edge; each lane handles a float4 (32*4 = 128 features)
// ---------------------------------------------------------------------------
__global__ void scatter1(const int* __restrict__ ei,
                         const float* __restrict__ dinv,
                         const float* __restrict__ h1,
                         float* __restrict__ agg1) {
    const int wid  = (blockIdx.x * blockDim.x + threadIdx.x) >> 5;
    const int lane = threadIdx.x & 31;
    if (wid >= NE) return;
    const int src = ei[wid];
    const int dst = ei[NE + wid];
    const float norm = dinv[src] * dinv[dst];
    const float4 v = ((const float4*)(h1 + (size_t)src * FHID))[lane];
    float* out = agg1 + (size_t)dst * FHID + lane * 4;
    atomicAdd(out + 0, v.x * norm);
    atomicAdd(out + 1, v.y * norm);
    atomicAdd(out + 2, v.z * norm);
    atomicAdd(out + 3, v.w * norm);
}

// ---------------------------------------------------------------------------
// 4) fused self-loop + bias + ReLU (in place on agg1):
//    h1post = relu(agg1 + h1 * dinv[n]^2 + b1)
// ---------------------------------------------------------------------------
__global__ void relu_self(const float* __restrict__ dinv,
                          const float* __restrict__ h1,
                          const float* __restrict__ b1,
                          float* __restrict__ agg1) {
    const size_t total4 = (size_t)NND * FHID / 4;
    const size_t stride = (size_t)gridDim.x * blockDim.x;
    for (size_t i = (size_t)blockIdx.x * blockDim.x + threadIdx.x; i < total4; i += stride) {
        const size_t e = i * 4;
        const int n = (int)(e / FHID);
        const int f = (int)(e % FHID);
        const float s = dinv[n];
        const float s2 = s * s;
        const float4 a  = ((const float4*)agg1)[i];
        const float4 hv = ((const float4*)h1)[i];
        float4 r;
        r.x = fmaxf(a.x + hv.x * s2 + b1[f + 0], 0.f);
        r.y = fmaxf(a.y + hv.y * s2 + b1[f + 1], 0.f);
        r.z = fmaxf(a.z + hv.z * s2 + b1[f + 2], 0.f);
        r.w = fmaxf(a.w + hv.w * s2 + b1[f + 3], 0.f);
        ((float4*)agg1)[i] = r;
    }
}

// ---------------------------------------------------------------------------
// 5) GEMM2: h2[N,2] = h1post[N,128] @ W2[128,2]
//    output is only 2 wide -> streaming dot products; W2 staged in LDS
// ---------------------------------------------------------------------------
__global__ void gemm2_kernel(const float* __restrict__ h1p,
                             const float* __restrict__ W2,
                             float* __restrict__ h2) {
    __shared__ float w2s[FHID * FOUT];
    if (threadIdx.x < FHID * FOUT) w2s[threadIdx.x] = W2[threadIdx.x];
    __syncthreads();

    const int n = blockIdx.x * blockDim.x + threadIdx.x;
    if (n >= NND) return;
    const float4* row = (const float4*)(h1p + (size_t)n * FHID);
    float acc0 = 0.f, acc1 = 0.f;
#pragma unroll 8
    for (int k4 = 0; k4 < FHID / 4; ++k4) {
        const float4 v = row[k4];
        const int k = k4 * 4;
        acc0 += v.x * w2s[(k + 0) * FOUT + 0] + v.y * w2s[(k + 1) * FOUT + 0]
              + v.z * w2s[(k + 2) * FOUT + 0] + v.w * w2s[(k + 3) * FOUT + 0];
        acc1 += v.x * w2s[(k + 0) * FOUT + 1] + v.y * w2s[(k + 1) * FOUT + 1]
              + v.z * w2s[(k + 2) * FOUT + 1] + v.w * w2s[(k + 3) * FOUT + 1];
    }
    h2[(size_t)n * FOUT + 0] = acc0;
    h2[(size_t)n * FOUT + 1] = acc1;
}

// ---------------------------------------------------------------------------
// 6) edge scatter layer 2 (2 features): agg2[dst,:] += h2[src,:] * norm
// ---------------------------------------------------------------------------
__global__ void scatter2(const int* __restrict__ ei,
                         const float* __restrict__ dinv,
                         const float* __restrict__ h2,
                         float* __restrict__ agg2) {
    const int e = blockIdx.x * blockDim.x + threadIdx.x;
    if (e >= NE) return;
    const int src = ei[e];
    const int dst = ei[NE + e];
    const float norm = dinv[src] * dinv[dst];
    atomicAdd(&agg2[(size_t)dst * FOUT + 0], h2[(size_t)src * FOUT + 0] * norm);
    atomicAdd(&agg2[(size_t)dst * FOUT + 1], h2[(size_t)src * FOUT + 1] * norm);
}

// ---------------------------------------------------------------------------
// 7) final epilogue: out = agg2 + h2 * dinv^2 + b2
// ---------------------------------------------------------------------------
__global__ void final_kernel(const float* __restrict__ dinv,
                             const float* __restrict__ h2,
                             const float* __restrict__ agg2,
                             const float* __restrict__ b2,
                             float* __restrict__ out) {
    const int n = blockIdx.x * blockDim.x + threadIdx.x;
    if (n >= NND) return;
    const float s = dinv[n];
    const float s2 = s * s;
    out[(size_t)n * FOUT + 0] = agg2[(size_t)n * FOUT + 0] + h2[(size_t)n * FOUT + 0] * s2 + b2[0];
    out[(size_t)n * FOUT + 1] = agg2[(size_t)n * FOUT + 1] + h2[(size_t)n * FOUT + 1] * s2 + b2[1];
}

// ---------------------------------------------------------------------------
// host launch
// ---------------------------------------------------------------------------
extern "C" void kernel_launch(void* const* d_in, const int* in_sizes, int n_in,
                              void* d_out, int out_size, void* d_ws, size_t ws_size,
                              hipStream_t stream) {
    const float* x  = (const float*)d_in[0];   // [N,165]
    const int*   ei = (const int*)d_in[1];     // [2,E]
    const float* W1 = (const float*)d_in[2];   // [165,128]
    const float* b1 = (const float*)d_in[3];   // [128]
    const float* W2 = (const float*)d_in[4];   // [128,2]
    const float* b2 = (const float*)d_in[5];   // [2]
    float* out = (float*)d_out;                // [N,2]

    // workspace layout (floats): [dinv N][agg1 N*128][agg2 N*2][h1 N*128][h2 N*2]
    float* ws   = (float*)d_ws;
    float* dinv = ws;
    float* agg1 = ws + (size_t)NND;
    float* agg2 = ws + (size_t)NND * 129;
    float* h1   = ws + (size_t)NND * 131;
    float* h2   = ws + (size_t)NND * 259;

    // zero deg + agg1 + agg2 (contiguous): N*131 floats
    const size_t zero4 = (size_t)NND * 131 / 4;
    zero_f4<<<2048, 256, 0, stream>>>((float4*)ws, zero4);

    degree_kernel<<<(NE + 255) / 256, 256, 0, stream>>>(ei, dinv);
    dinv_kernel<<<(NND + 255) / 256, 256, 0, stream>>>(dinv);

    // GEMM1: 12500*8 = 100000 tiles, 4 waves/block -> 25000 blocks of 128
    gemm1_wmma<<<25000, 128, 0, stream>>>(x, W1, h1);

    // scatter layer 1: one wave per edge, 8 waves per 256-thread block
    scatter1<<<(NE + 7) / 8, 256, 0, stream>>>(ei, dinv, h1, agg1);

    relu_self<<<2048, 256, 0, stream>>>(dinv, h1, b1, agg1);

    gemm2_kernel<<<(NND + 255) / 256, 256, 0, stream>>>(agg1, W2, h2);

    scatter2<<<(NE + 255) / 256, 256, 0, stream>>>(ei, dinv, h2, agg2);

    final_kernel<<<(NND + 255) / 256, 256, 0, stream>>>(dinv, h2, agg2, b2, out);
}